// Net_87686052315847
// MI455X (gfx1250) — compile-verified
//
#include <hip/hip_runtime.h>
#include <math.h>

typedef __attribute__((ext_vector_type(2))) float v2f;
typedef __attribute__((ext_vector_type(8))) float v8f;

// ---------------------------------------------------------------------------
// Pass 0: init deg[n] = 1.0 (self loop), zero per-graph accumulators
// ---------------------------------------------------------------------------
__global__ void k_init(float* __restrict__ deg, float* __restrict__ S,
                       float* __restrict__ cnt, int N, int G) {
    int i = blockIdx.x * blockDim.x + threadIdx.x;
    if (i < N) deg[i] = 1.0f;
    if (i < G) { S[i] = 0.0f; cnt[i] = 0.0f; }
}

// ---------------------------------------------------------------------------
// Pass 1: deg[dst[e]] += 1  (incoming-degree, matches segment_sum over dst)
// ---------------------------------------------------------------------------
__global__ void k_deg(const int* __restrict__ dst, float* __restrict__ deg, int E) {
    int e = blockIdx.x * blockDim.x + threadIdx.x;
    if (e < E) atomicAdd(&deg[dst[e]], 1.0f);
}

// ---------------------------------------------------------------------------
// Pass 2: dinv[n] = rsqrt(deg[n]) (in place); s[n] = x[n]*dinv[n]^2 (self loop)
// ---------------------------------------------------------------------------
__global__ void k_dinv_self(const float* __restrict__ x, float* __restrict__ deg,
                            float* __restrict__ s, int N) {
    int n = blockIdx.x * blockDim.x + threadIdx.x;
    if (n < N) {
        float di = rsqrtf(deg[n]);
        deg[n] = di;                 // deg buffer now holds dinv
        s[n] = x[n] * di * di;       // self-loop message (scalar, rank-1 collapse)
    }
}

// ---------------------------------------------------------------------------
// Pass 3: scalar edge scatter: s[dst] += x[src]*dinv[src]*dinv[dst]
// (the 64-wide message collapses to a scalar because h = x ⊗ W1 is rank-1)
// ---------------------------------------------------------------------------
__global__ void k_msg(const int* __restrict__ src, const int* __restrict__ dst,
                      const float* __restrict__ x, const float* __restrict__ dinv,
                      float* __restrict__ s, int E) {
    int e = blockIdx.x * blockDim.x + threadIdx.x;
    if (e < E) {
        int si = src[e];
        int di = dst[e];
        atomicAdd(&s[di], x[si] * dinv[si] * dinv[di]);
    }
}

// ---------------------------------------------------------------------------
// Pass 4: per-graph pooling. batch is sorted -> contiguous segments, so do a
// wave32 segmented shfl_down reduction; only segment leaders issue atomics.
// ---------------------------------------------------------------------------
__global__ void k_pool(const float* __restrict__ s, const int* __restrict__ batch,
                       float* __restrict__ S, float* __restrict__ cnt, int N) {
    int n = blockIdx.x * blockDim.x + threadIdx.x;
    int lane = threadIdx.x & 31;
    int g = -1;
    float v = 0.0f, c = 0.0f;
    if (n < N) { g = batch[n]; v = s[n]; c = 1.0f; }
    // segmented reduce over contiguous equal-g runs within the wave
    for (int off = 1; off < 32; off <<= 1) {
        int   go = __shfl_down(g, off, 32);
        float vo = __shfl_down(v, off, 32);
        float co = __shfl_down(c, off, 32);
        if (lane + off < 32 && go == g) { v += vo; c += co; }
    }
    int gprev = __shfl_up(g, 1, 32);
    bool leader = (lane == 0) || (gprev != g);
    if (leader && n < N) {
        atomicAdd(&S[g], v);
        atomicAdd(&cnt[g], c);
    }
}

// ---------------------------------------------------------------------------
// Pass 5: m[g] = S[g] / max(cnt[g], 1)   (in place in S)
// ---------------------------------------------------------------------------
__global__ void k_mean(float* __restrict__ S, const float* __restrict__ cnt, int G) {
    int g = blockIdx.x * blockDim.x + threadIdx.x;
    if (g < G) S[g] = S[g] / fmaxf(cnt[g], 1.0f);
}

// ---------------------------------------------------------------------------
// Pass 6: classifier via V_WMMA_F32_16X16X4_F32 + log_softmax.
// Per wave: 16-graph x 16-class tile (classes 10..15 zero padded).
//   A[row][k] = m[gbase+row]*W1[k] + b1[k]   (mean matrix, built on the fly)
//   B[k][col] = W2[k][col]  (staged zero-padded + transposed in LDS so the
//                            inner loop has no predication / EXEC churn and
//                            each B pair is one aligned ds_load_b64)
// A layout (32-bit 16x4): lanes 0-15 = rows, half = lane>>4 selects K pair
//   a.x = A[row][kb+2*half], a.y = A[row][kb+2*half+1]
// B layout (32-bit 4x16): b.x = B[kb+2*half][col], b.y = B[kb+2*half+1][col]
// D layout: c[r] in lane L = D[r + 8*(L>>4)][L&15]
// ---------------------------------------------------------------------------
__global__ void k_cls(const float* __restrict__ m, const float* __restrict__ W1,
                      const float* __restrict__ b1, const float* __restrict__ W2,
                      const float* __restrict__ b2, float* __restrict__ out, int G) {
    __shared__ float W2t[16 * 64];   // [col][k], cols 10..15 = 0
    __shared__ float b2s[16];        // classes 10..15 = 0

    // cooperative zero-padded transpose fill (1024 + 16 elements, 256 threads)
    for (int i = threadIdx.x; i < 16 * 64; i += blockDim.x) {
        int col = i >> 6, k = i & 63;
        W2t[i] = (col < 10) ? W2[k * 10 + col] : 0.0f;
    }
    if (threadIdx.x < 16) b2s[threadIdx.x] = (threadIdx.x < 10) ? b2[threadIdx.x] : 0.0f;
    __syncthreads();

    int wave  = threadIdx.x >> 5;
    int lane  = threadIdx.x & 31;
    int tile  = blockIdx.x * (blockDim.x >> 5) + wave;
    int gbase = tile * 16;
    if (gbase >= G) return;            // grid sized so this never trips (EXEC all 1s)

    int  half  = lane >> 4;
    int  rc    = lane & 15;            // row for A-frag, col for B/C/D frags
    bool valid = rc < 10;

    float mrow = m[gbase + rc];
    const float* w2row = &W2t[rc * 64];

    v8f c = {};
    #pragma unroll
    for (int kb = 0; kb < 64; kb += 4) {
        int k0 = kb + 2 * half;
        v2f a;
        a.x = fmaf(mrow, W1[k0],     b1[k0]);
        a.y = fmaf(mrow, W1[k0 + 1], b1[k0 + 1]);
        v2f b = *(const v2f*)&w2row[k0];   // 8B-aligned ds_load_b64, pad cols are 0
        c = __builtin_amdgcn_wmma_f32_16x16x4_f32(
                /*neg_a=*/false, a, /*neg_b=*/false, b,
                /*c_mod=*/(short)0, c, /*reuse_a=*/false, /*reuse_b=*/false);
    }

    float bias = b2s[rc];              // 0 for pad classes
    #pragma unroll
    for (int r = 0; r < 8; ++r) {
        float lg = c[r] + bias;
        // max over the 10 valid classes (16-lane group; xor masks 1..8 stay in-half)
        float mx = valid ? lg : -3.402823466e38f;
        for (int off = 8; off; off >>= 1) mx = fmaxf(mx, __shfl_xor(mx, off, 32));
        float ex = valid ? expf(lg - mx) : 0.0f;
        float sm = ex;
        for (int off = 8; off; off >>= 1) sm += __shfl_xor(sm, off, 32);
        float lse = mx + logf(sm);
        if (valid) out[(gbase + r + 8 * half) * 10 + rc] = lg - lse;
    }
}

// ---------------------------------------------------------------------------
extern "C" void kernel_launch(void* const* d_in, const int* in_sizes, int n_in,
                              void* d_out, int out_size, void* d_ws, size_t ws_size,
                              hipStream_t stream) {
    const float* x     = (const float*)d_in[0];
    const int*   ei    = (const int*)  d_in[1];
    const int*   batch = (const int*)  d_in[2];
    const float* W1    = (const float*)d_in[3];
    const float* b1    = (const float*)d_in[4];
    const float* W2    = (const float*)d_in[5];
    const float* b2    = (const float*)d_in[6];
    float*       out   = (float*)d_out;

    const int N = in_sizes[0];          // 307200
    const int E = in_sizes[1] / 2;      // 5734400
    const int G = out_size / 10;        // 4096

    const int* src = ei;
    const int* dst = ei + E;

    // workspace layout (floats): deg/dinv[N] | s[N] | S->mean[G] | cnt[G]
    float* ws   = (float*)d_ws;
    float* deg  = ws;                   // becomes dinv after pass 2
    float* s    = ws + N;
    float* Sg   = ws + 2 * (size_t)N;   // becomes mean after pass 5
    float* cntg = Sg + G;

    const int T = 256;
    int gN = (N + T - 1) / T;
    int gE = (E + T - 1) / T;
    int gG = (G + T - 1) / T;

    k_init<<<gN, T, 0, stream>>>(deg, Sg, cntg, N, G);
    k_deg<<<gE, T, 0, stream>>>(dst, deg, E);
    k_dinv_self<<<gN, T, 0, stream>>>(x, deg, s, N);
    k_msg<<<gE, T, 0, stream>>>(src, dst, x, deg, s, E);
    k_pool<<<gN, T, 0, stream>>>(s, batch, Sg, cntg, N);
    k_mean<<<gG, T, 0, stream>>>(Sg, cntg, G);

    // 256 tiles of 16 graphs; 8 waves/block -> 32 blocks, every wave has a tile
    int tiles = G / 16;
    int wavesPerBlock = T / 32;
    int gC = (tiles + wavesPerBlock - 1) / wavesPerBlock;
    k_cls<<<gC, T, 0, stream>>>(Sg, W1, b1, W2, b2, out, G);
}